// EMLTree_24661702213970
// MI455X (gfx1250) — compile-verified
//
#include <hip/hip_runtime.h>
#include <math.h>

// ---------------------------------------------------------------------------
// EML tree evaluator, MI455X (gfx1250, wave32).
// Trans-bound kernel: 127 nodes/element, 2 transcendentals per node.
// Weights: softmax'ed ONCE by a tiny prep kernel into d_ws, l-row prescaled by
// log2(e) so main loop uses raw v_exp_f32 / v_log_f32.
// Workspace layout: ws[node*6] = { a_l*log2e, b_l*log2e, c_l*log2e,
//                                  a_r,       b_r,       c_r }
// Node order: level0 g=0..63 (c=0), level1 g=0..31, ..., level6 g=0. 127 nodes.
//
// CDNA5-specific paths used:
//   - gfx1250 async global->LDS gather (global_load_async_to_lds_b32 +
//     s_wait_asynccnt) in the one-shot prep kernel (zero hot-path cost)
//   - global_prefetch_b8 on the streaming input
//   - native v_exp_f32 / v_log_f32 / v_med3_f32 via builtins
// ---------------------------------------------------------------------------

#define EML_LOG2E 1.44269504088896340736f
#define EML_LN2   0.69314718055994530942f
#define EML_CLIP  (8.0f * EML_LOG2E)

#if defined(__HIP_DEVICE_COMPILE__) && \
    __has_builtin(__builtin_amdgcn_global_load_async_to_lds_b32)
#define EML_HAS_ASYNC_LDS 1
#endif

__device__ __forceinline__ void eml_softmax_row(const float* p, int nc, float* w) {
    float m = p[0];
    for (int k = 1; k < nc; ++k) m = fmaxf(m, p[k]);
    float s = 0.0f;
    for (int k = 0; k < nc; ++k) { w[k] = expf(p[k] - m); s += w[k]; }
    float inv = 1.0f / s;
    for (int k = 0; k < nc; ++k) w[k] *= inv;
}

// One-time: 127 threads, each softmaxes one tree node (both rows) into ws.
// Logit rows are gathered into LDS via the gfx1250 async global->LDS path.
__global__ void eml_prep_kernel(const float* __restrict__ l0,
                                const float* __restrict__ l1,
                                const float* __restrict__ l2,
                                const float* __restrict__ l3,
                                const float* __restrict__ l4,
                                const float* __restrict__ l5,
                                const float* __restrict__ l6,
                                float* __restrict__ ws) {
    __shared__ float slog[128 * 6];
    int n = threadIdx.x;
    if (n >= 127) return;

    const float* base;
    int g, nc;
    if      (n <  64) { base = l0; g = n;       nc = 2; }
    else if (n <  96) { base = l1; g = n -  64; nc = 3; }
    else if (n < 112) { base = l2; g = n -  96; nc = 3; }
    else if (n < 120) { base = l3; g = n - 112; nc = 3; }
    else if (n < 124) { base = l4; g = n - 120; nc = 3; }
    else if (n < 126) { base = l5; g = n - 124; nc = 3; }
    else              { base = l6; g = 0;       nc = 3; }

    const float* rowl = base + (size_t)g * 2 * nc;
    const float* rowr = rowl + nc;
    float* sl = &slog[n * 6];      // [0..2] = l-row logits, [3..5] = r-row

#if defined(EML_HAS_ASYNC_LDS)
    // Per-lane async gather: global logits -> LDS (tracked by ASYNCcnt).
    for (int k = 0; k < nc; ++k) {
        __builtin_amdgcn_global_load_async_to_lds_b32(
            (__attribute__((address_space(1))) int*)(rowl + k),
            (__attribute__((address_space(3))) int*)(sl + k), 0, 0);
        __builtin_amdgcn_global_load_async_to_lds_b32(
            (__attribute__((address_space(1))) int*)(rowr + k),
            (__attribute__((address_space(3))) int*)(sl + 3 + k), 0, 0);
    }
#if __has_builtin(__builtin_amdgcn_s_wait_asynccnt)
    __builtin_amdgcn_s_wait_asynccnt(0);
#else
    asm volatile("s_wait_asynccnt 0x0" ::: "memory");
#endif
#else
    for (int k = 0; k < nc; ++k) { sl[k] = rowl[k]; sl[3 + k] = rowr[k]; }
#endif

    float wl[3] = {0.f, 0.f, 0.f}, wr[3] = {0.f, 0.f, 0.f};
    eml_softmax_row(sl,     nc, wl);
    eml_softmax_row(sl + 3, nc, wr);

    float* o = ws + (size_t)n * 6;
    o[0] = wl[0] * EML_LOG2E;
    o[1] = wl[1] * EML_LOG2E;
    o[2] = (nc == 3) ? wl[2] * EML_LOG2E : 0.0f;
    o[3] = wr[0];
    o[4] = wr[1];
    o[5] = (nc == 3) ? wr[2] : 0.0f;
}

// Main kernel: one element per lane, fully unrolled 127-node tree.
// Weight loads are uniform-address + compile-time offsets -> batched s_load_bN
// through the scalar/constant cache, co-executing with the VALU/TRANS work.
__global__ __launch_bounds__(256)
void eml_main_kernel(const float* __restrict__ x,
                     const float* __restrict__ w,
                     float* __restrict__ out, int n) {
    int i = blockIdx.x * blockDim.x + threadIdx.x;
    if (i >= n) return;
    __builtin_prefetch(&x[i], 0, 1);       // gfx1250 global_prefetch_b8
    const float xv = x[i];

    float cur[64];

    // Level 0: 64 nodes, no feedback term (c == 0).
    #pragma unroll
    for (int g = 0; g < 64; ++g) {
        const float* wp = w + 6 * g;
        float l = fmaf(wp[1], xv, wp[0]);                    // already *log2e
        l = __builtin_amdgcn_fmed3f(l, -EML_CLIP, EML_CLIP); // clamp
        float e = __builtin_amdgcn_exp2f(l);                 // v_exp_f32
        float r = fmaf(wp[4], xv, wp[3]);
        float lg = __builtin_amdgcn_logf(fabsf(r));          // v_log_f32(|r|)
        cur[g] = fmaf(-EML_LN2, lg, e);                      // exp(l) - ln|r|
    }

    // Levels 1..6: in-place pairwise fold (safe ascending: g <= 2g).
    int base = 64;
    #pragma unroll
    for (int lev = 1; lev <= 6; ++lev) {
        const int width = 64 >> lev;
        #pragma unroll
        for (int g = 0; g < width; ++g) {
            const float* wp = w + 6 * (base + g);
            const float fl = cur[2 * g];
            const float fr = cur[2 * g + 1];
            float l = fmaf(wp[1], xv, wp[0]);
            l = fmaf(wp[2], fl, l);
            l = __builtin_amdgcn_fmed3f(l, -EML_CLIP, EML_CLIP);
            float e = __builtin_amdgcn_exp2f(l);
            float r = fmaf(wp[4], xv, wp[3]);
            r = fmaf(wp[5], fr, r);
            float lg = __builtin_amdgcn_logf(fabsf(r));
            cur[g] = fmaf(-EML_LN2, lg, e);
        }
        base += width;
    }

    out[i] = cur[0];
}

extern "C" void kernel_launch(void* const* d_in, const int* in_sizes, int n_in,
                              void* d_out, int out_size, void* d_ws, size_t ws_size,
                              hipStream_t stream) {
    const float* x  = (const float*)d_in[0];
    const float* l0 = (const float*)d_in[1];
    const float* l1 = (const float*)d_in[2];
    const float* l2 = (const float*)d_in[3];
    const float* l3 = (const float*)d_in[4];
    const float* l4 = (const float*)d_in[5];
    const float* l5 = (const float*)d_in[6];
    const float* l6 = (const float*)d_in[7];
    float* ws  = (float*)d_ws;   // needs 127*6*4 = 3048 bytes
    float* out = (float*)d_out;

    // 1) Softmax all 127 nodes once (prescaled weights) into workspace.
    eml_prep_kernel<<<1, 128, 0, stream>>>(l0, l1, l2, l3, l4, l5, l6, ws);

    // 2) Elementwise tree evaluation: 1 element / lane, wave32.
    const int n = in_sizes[0];           // B = 1048576
    const int threads = 256;             // 8 wave32 per block
    const int blocks = (n + threads - 1) / threads;
    eml_main_kernel<<<blocks, threads, 0, stream>>>(x, ws, out, n);
}